// LabelFusionForToken_4260607558419
// MI455X (gfx1250) — compile-verified
//
#include <hip/hip_runtime.h>
#include <hip/hip_bf16.h>

typedef float v2f __attribute__((ext_vector_type(2)));
typedef float v4f __attribute__((ext_vector_type(4)));
typedef float v8f __attribute__((ext_vector_type(8)));

#define KH 768   // hidden dim
#define KC 24    // classes
#define KN 16    // label tokens per class
#define KM 4096  // BS*SEQ
#define KB 32    // k-chunk staged in LDS
#define WPAD 36  // LDS row stride (floats): 16B-aligned rows, conflict-free frags

// ---------------------------------------------------------------------------
// C[M,768] = A[M,768] @ W[768,768]^T   (f32 WMMA 16x16x4)
// Block = 8 waves = 256 threads, covers 256(M) x 64(N).
// Wave computes 32(M) x 64(N): 2 M-tiles x 4 N-tiles = 8 accumulators.
// W strip (64 rows x KB cols) is staged per block into LDS with coalesced
// float4 loads (all waves share the same j-strip), then B-fragments are
// conflict-free DS float2 reads. Inner step: 2 global + 4 DS loads -> 8 WMMA.
// Frags: A 16x4 MxK -> lane L: A[m0+L%16, k+2*(L/16)+{0,1}]
//        B 4x16 KxN -> B[k,n]=W[j0+n,k] -> lane L: W row j0+L%16 (via LDS)
// C/D:   VGPR r, lane L -> C[m0 + r + 8*(L/16), j0 + L%16]
// ---------------------------------------------------------------------------
__global__ __launch_bounds__(256) void gemm_xWt(const float* __restrict__ A,
                                                const float* __restrict__ W,
                                                float* __restrict__ Cm,
                                                int M) {
  __shared__ float ldsW[64][WPAD];  // 9216 B
  const int lane = threadIdx.x & 31;
  const int wave = threadIdx.x >> 5;
  const int m0 = blockIdx.x * 256 + wave * 32;
  const int j0 = blockIdx.y * 64;
  const int lm = lane & 15;
  const int hi = lane >> 4;
  const int koff = hi * 2;
  const bool active = (m0 < M);

  const float* arow0 = A + (size_t)(m0 + lm) * KH + koff;
  const float* arow1 = arow0 + 16 * KH;

  // staging map: thread t -> W row j0 + t/4, cols (t%4)*8 .. +7 (two float4)
  const int srow = threadIdx.x >> 2;
  const int scol = (threadIdx.x & 3) * 8;
  const float* wsrc = W + (size_t)(j0 + srow) * KH + scol;

  v8f acc[2][4] = {};

  for (int kb = 0; kb < KH; kb += KB) {
    __syncthreads();
    *(v4f*)&ldsW[srow][scol]     = *(const v4f*)(wsrc + kb);
    *(v4f*)&ldsW[srow][scol + 4] = *(const v4f*)(wsrc + kb + 4);
    __syncthreads();
    if (kb + KB < KH) __builtin_prefetch(wsrc + kb + KB, 0, 3);

    if (active) {
#pragma unroll
      for (int kk = 0; kk < KB; kk += 4) {
        v2f a0 = *(const v2f*)(arow0 + kb + kk);
        v2f a1 = *(const v2f*)(arow1 + kb + kk);
#pragma unroll
        for (int t = 0; t < 4; ++t) {
          v2f b = *(const v2f*)&ldsW[t * 16 + lm][kk + koff];
          acc[0][t] = __builtin_amdgcn_wmma_f32_16x16x4_f32(false, a0, false, b, (short)0, acc[0][t], false, false);
          acc[1][t] = __builtin_amdgcn_wmma_f32_16x16x4_f32(false, a1, false, b, (short)0, acc[1][t], false, false);
        }
      }
    }
  }

  if (active) {
    float* crow = Cm + (size_t)(m0 + (hi << 3)) * KH + j0 + lm;
#pragma unroll
    for (int mt = 0; mt < 2; ++mt) {
#pragma unroll
      for (int r = 0; r < 8; ++r) {
#pragma unroll
        for (int t = 0; t < 4; ++t)
          crow[(mt * 16 + r) * KH + t * 16] = acc[mt][t][r];
      }
    }
  }
}

// ---------------------------------------------------------------------------
// Fused attention + output. One wave per (16-token tile, class c):
//   scores(16x16) = tf_tile @ lf_c^T            (192 WMMA k-steps)
//   softmax over n=16 (mask bias) via width-16 shfl_xor butterflies
//   attn staged to LDS, reloaded as A-fragments (K=16 => 4 frags)
//   48 j-tiles: acc = attn @ lfW_c (4 WMMA), out = tanh(acc + tfW_tile)
// ---------------------------------------------------------------------------
__global__ __launch_bounds__(256) void attn_out(const float* __restrict__ tf,
                                                const float* __restrict__ lf,
                                                const float* __restrict__ tfW,
                                                const float* __restrict__ lfW,
                                                const float* __restrict__ mask,
                                                float* __restrict__ out) {
  __shared__ float attn_lds[8][16][16];  // per-wave 16x16 attn tile (8 KB)
  const int lane = threadIdx.x & 31;
  const int wave = threadIdx.x >> 5;
  const int p = blockIdx.x * 8 + wave;   // 0 .. 6143
  const int c = p % KC;
  const int bm0 = (p / KC) * 16;
  const int lm = lane & 15;
  const int hi = lane >> 4;
  const int koff = hi * 2;

  // ---- scores = tf_tile(16x768) @ lf_c(16x768)^T ----
  const float* arow = tf + (size_t)(bm0 + lm) * KH + koff;
  const float* brow = lf + (size_t)(c * KN + lm) * KH + koff;
  v8f s = {};
  for (int k = 0; k < KH; k += 4) {
    v2f a = *(const v2f*)(arow + k);
    v2f b = *(const v2f*)(brow + k);
    s = __builtin_amdgcn_wmma_f32_16x16x4_f32(false, a, false, b, (short)0, s, false, false);
  }

  // ---- mask + softmax over n (n = lm; rows r and r+8 in the two halves) ----
  const float bias = (1.0f - mask[c * KN + lm]) * -10000.0f;
#pragma unroll
  for (int r = 0; r < 8; ++r) {
    float v = s[r] + bias;
    float mx = v;
    for (int off = 8; off; off >>= 1) mx = fmaxf(mx, __shfl_xor(mx, off, 16));
    float e = __expf(v - mx);
    float sum = e;
    for (int off = 8; off; off >>= 1) sum += __shfl_xor(sum, off, 16);
    attn_lds[wave][r + (hi << 3)][lm] = e / sum;
  }

  // ---- reload attn as A-fragments: K=16 -> 4 fragments of K=4 ----
  v2f afrag[4];
#pragma unroll
  for (int kk = 0; kk < 4; ++kk) {
    afrag[kk].x = attn_lds[wave][lm][4 * kk + koff + 0];
    afrag[kk].y = attn_lds[wave][lm][4 * kk + koff + 1];
  }

  // ---- out[bm, c, :] = tanh(tfW[bm,:] + attn @ lfW_c) ----
  const float* lfWc = lfW + (size_t)c * KN * KH;
  const int mrow = bm0 + (hi << 3);
  for (int j0 = 0; j0 < KH; j0 += 16) {
    v8f acc = {};
#pragma unroll
    for (int kk = 0; kk < 4; ++kk) {
      const int k = 4 * kk + koff;
      v2f b;
      b.x = lfWc[(k + 0) * KH + j0 + lm];
      b.y = lfWc[(k + 1) * KH + j0 + lm];
      acc = __builtin_amdgcn_wmma_f32_16x16x4_f32(false, afrag[kk], false, b, (short)0, acc, false, false);
    }
#pragma unroll
    for (int r = 0; r < 8; ++r) {
      const int row = mrow + r;
      const int col = j0 + lm;
      float v = tanhf(acc[r] + tfW[(size_t)row * KH + col]);
      out[((size_t)row * KC + c) * KH + col] = v;
    }
  }
}

extern "C" void kernel_launch(void* const* d_in, const int* in_sizes, int n_in,
                              void* d_out, int out_size, void* d_ws, size_t ws_size,
                              hipStream_t stream) {
  const float* token = (const float*)d_in[0];  // (8,512,768)
  const float* label = (const float*)d_in[1];  // (24,16,768)
  const float* mask  = (const float*)d_in[2];  // (24,16)
  const float* W1    = (const float*)d_in[3];
  const float* W2    = (const float*)d_in[4];
  const float* Wout  = (const float*)d_in[5];
  float* out = (float*)d_out;

  float* ws  = (float*)d_ws;
  float* tf  = ws;                       // 4096*768
  float* lf  = tf  + (size_t)KM * KH;    //  384*768
  float* tfW = lf  + (size_t)KC * KN * KH;
  float* lfW = tfW + (size_t)KM * KH;    // total ~27.5 MB

  // tf = token @ W1^T ; lf = label @ W2^T
  gemm_xWt<<<dim3(16, 12), 256, 0, stream>>>(token, W1, tf, KM);
  gemm_xWt<<<dim3(2, 12), 256, 0, stream>>>(label, W2, lf, KC * KN);
  // tfW = tf @ Wout^T ; lfW = lf @ Wout^T   (Wout distributed over the sum)
  gemm_xWt<<<dim3(16, 12), 256, 0, stream>>>(tf, Wout, tfW, KM);
  gemm_xWt<<<dim3(2, 12), 256, 0, stream>>>(lf, Wout, lfW, KC * KN);
  // fused scores/softmax/weighted/tanh epilogue
  attn_out<<<dim3(768), 256, 0, stream>>>(tf, lf, tfW, lfW, mask, out);
}